// AttentiveGraphNet_52364241273391
// MI455X (gfx1250) — compile-verified
//
#include <hip/hip_runtime.h>
#include <cstddef>
#include <cstdint>

// ---------------------------------------------------------------------------
// AttentiveGraphNet forward for MI455X (gfx1250), fp32 with V_WMMA_F32_16X16X4_F32.
//
// Input pointer order assumption: harness flattens the setup_inputs() pytree the
// way jax.tree_util does (dict keys sorted recursively):
//   0:x 1:edge_index 2:edge_attr 3:batch
//   4:W_enc1 5:cheb_W 6:cheb_b 7:edge_W 8:edge_b
//   9..20:enc[0]{ff1_W,ff1_b,ff2_W,ff2_b,in_W,in_b,ln1.b,ln1.g,ln2.b,ln2.g,out_W,out_b}
//   21..32:enc[1]{same}
//   33:fc_W 34:fc_b 35:lin1_W 36:lin1_b 37:lin2_W 38:lin2_b
//   39..46:norm1..4 {b,g}
//   47..55:tc1{We,Wk,Wq,Wskip,Wv,bk,bq,bskip,bv}  56..64:tc2{same}
// Workspace requirement: ~95 MB.
// ---------------------------------------------------------------------------

#define NODES 4096
#define EDGES 65536
#define CIN_  64
#define HC_   128
#define EDIM_ 32
#define FF_   256
#define HEADS_ 4
#define HD_   32

#define CDIV(a,b) (((a)+(b)-1)/(b))

typedef float v2f __attribute__((ext_vector_type(2)));
typedef float v8f __attribute__((ext_vector_type(8)));

__device__ __forceinline__ v8f wmma4(v2f a, v2f b, v8f c) {
  // D(16x16,f32) = A(16x4,f32) * B(4x16,f32) + C
  return __builtin_amdgcn_wmma_f32_16x16x4_f32(false, a, false, b, (short)0, c, false, false);
}

// ---------------- generic WMMA GEMM: one wave per 16x16 C tile ----------------
// EPI bit0: +bias[col]   bit1: relu   bit2: +R[row,col]
template <int EPI>
__global__ __launch_bounds__(256) void gemm16(const float* __restrict__ A, int lda,
                                              const float* __restrict__ B, int ldb,
                                              const float* __restrict__ bias,
                                              const float* R,
                                              float* C, int ldc,
                                              int M, int Nc, int K) {
  int wid = (blockIdx.x * blockDim.x + threadIdx.x) >> 5;
  int tn_cnt = Nc >> 4;
  int tm = wid / tn_cnt;
  int tn = wid - tm * tn_cnt;
  if (tm >= (M >> 4)) return;            // wave-uniform: EXEC stays all-ones
  int lane = threadIdx.x & 31;
  int m  = lane & 15;
  int kh = (lane >> 4) << 1;             // 0 or 2 (A/B K-pair per half-wave)
  const float* Ap = A + (size_t)(tm * 16 + m) * lda;
  const float* Bp = B + tn * 16 + m;
  v8f acc = {0.f,0.f,0.f,0.f,0.f,0.f,0.f,0.f};
  for (int k = 0; k < K; k += 4) {
    v2f a, b;
    a.x = Ap[k + kh];
    a.y = Ap[k + kh + 1];
    b.x = Bp[(size_t)(k + kh) * ldb];
    b.y = Bp[(size_t)(k + kh + 1) * ldb];
    acc = wmma4(a, b, acc);
  }
  int col = tn * 16 + m;
#pragma unroll
  for (int i = 0; i < 8; i++) {
    int row = tm * 16 + ((lane >> 4) << 3) + i;   // C layout: vgpr i -> row (half*8+i)
    float v = acc[i];
    if (EPI & 1) v += bias[col];
    if (EPI & 4) v += R[(size_t)row * ldc + col];
    if (EPI & 2) v = fmaxf(v, 0.f);
    C[(size_t)row * ldc + col] = v;
  }
}

// ---------------- dense encoder attention (flash style, WMMA) ----------------
// qkv: [NODES, 384] (q|k|v, each [NODES,4,32]).  out: [NODES,128].
// One wave handles 16 queries of one head, streaming keys in 16-chunks.
__global__ __launch_bounds__(256) void attn_kernel(const float* __restrict__ qkv,
                                                   float* __restrict__ out) {
  __shared__ float pls[8 * 256];
  int lwave = threadIdx.x >> 5;
  int wave  = blockIdx.x * 8 + lwave;
  int head  = wave & (HEADS_ - 1);
  int qt    = wave >> 2;
  if (qt >= NODES / 16) return;
  int lane = threadIdx.x & 31;
  int m  = lane & 15;
  int kh = (lane >> 4) << 1;
  const int ld = 3 * HC_;
  const float* Q  = qkv + (size_t)(qt * 16) * ld + head * HD_;
  const float* Kp = qkv + HC_ + head * HD_;
  const float* Vp = qkv + 2 * HC_ + head * HD_;

  v2f aq[8];
#pragma unroll
  for (int kk = 0; kk < 8; kk++) {
    aq[kk].x = Q[(size_t)m * ld + kk * 4 + kh];
    aq[kk].y = Q[(size_t)m * ld + kk * 4 + kh + 1];
  }
  float Mx[8], Ls[8];
  v8f o0 = {0.f,0.f,0.f,0.f,0.f,0.f,0.f,0.f};
  v8f o1 = {0.f,0.f,0.f,0.f,0.f,0.f,0.f,0.f};
#pragma unroll
  for (int i = 0; i < 8; i++) { Mx[i] = -1e30f; Ls[i] = 0.f; }
  float* P = pls + lwave * 256;

  for (int kt = 0; kt < NODES; kt += 16) {
    // S = Q * K^T   (A = Q 16x32, B = K^T 32x16; lane's B column = key kt+m)
    v8f s = {0.f,0.f,0.f,0.f,0.f,0.f,0.f,0.f};
#pragma unroll
    for (int kk = 0; kk < 8; kk++) {
      v2f b;
      b.x = Kp[(size_t)(kt + m) * ld + kk * 4 + kh];
      b.y = Kp[(size_t)(kt + m) * ld + kk * 4 + kh + 1];
      s = wmma4(aq[kk], b, s);
    }
    float tm_[8], rs[8];
#pragma unroll
    for (int i = 0; i < 8; i++) { s[i] *= 0.17677669529663687f; tm_[i] = s[i]; }
    // row reductions: rows live entirely within one 16-lane half
#pragma unroll
    for (int off = 1; off < 16; off <<= 1)
#pragma unroll
      for (int i = 0; i < 8; i++) tm_[i] = fmaxf(tm_[i], __shfl_xor(tm_[i], off));
#pragma unroll
    for (int i = 0; i < 8; i++) {
      float nm = fmaxf(Mx[i], tm_[i]);
      float sc = __expf(Mx[i] - nm);
      Mx[i] = nm;
      o0[i] *= sc; o1[i] *= sc; Ls[i] *= sc;
      s[i] = __expf(s[i] - nm);
      rs[i] = s[i];
    }
#pragma unroll
    for (int off = 1; off < 16; off <<= 1)
#pragma unroll
      for (int i = 0; i < 8; i++) rs[i] += __shfl_xor(rs[i], off);
#pragma unroll
    for (int i = 0; i < 8; i++) Ls[i] += rs[i];
    // C-layout -> LDS -> A-layout relayout of P for the P*V WMMA
#pragma unroll
    for (int i = 0; i < 8; i++) P[(((lane >> 4) << 3) + i) * 16 + m] = s[i];
#pragma unroll
    for (int kk = 0; kk < 4; kk++) {
      v2f a, b0, b1;
      a.x = P[m * 16 + kk * 4 + kh];
      a.y = P[m * 16 + kk * 4 + kh + 1];
      b0.x = Vp[(size_t)(kt + kk * 4 + kh) * ld + m];
      b0.y = Vp[(size_t)(kt + kk * 4 + kh + 1) * ld + m];
      o0 = wmma4(a, b0, o0);
      b1.x = Vp[(size_t)(kt + kk * 4 + kh) * ld + 16 + m];
      b1.y = Vp[(size_t)(kt + kk * 4 + kh + 1) * ld + 16 + m];
      o1 = wmma4(a, b1, o1);
    }
  }
#pragma unroll
  for (int i = 0; i < 8; i++) {
    int r = qt * 16 + ((lane >> 4) << 3) + i;
    float inv = 1.0f / Ls[i];
    out[(size_t)r * HC_ + head * HD_ + m]      = o0[i] * inv;
    out[(size_t)r * HC_ + head * HD_ + 16 + m] = o1[i] * inv;
  }
}

// ---------------- LayerNorm: one wave per 128-wide row ----------------
__global__ __launch_bounds__(256) void ln_kernel(const float* __restrict__ X,
                                                 const float* Rsd,
                                                 const float* __restrict__ g,
                                                 const float* __restrict__ b,
                                                 float* out, int rows) {
  int row = (blockIdx.x * blockDim.x + threadIdx.x) >> 5;
  if (row >= rows) return;
  int lane = threadIdx.x & 31;
  float v[4], s = 0.f, s2 = 0.f;
#pragma unroll
  for (int j = 0; j < 4; j++) {
    int c = lane + j * 32;
    float t = X[(size_t)row * HC_ + c];
    if (Rsd) t += Rsd[(size_t)row * HC_ + c];
    v[j] = t; s += t; s2 += t * t;
  }
#pragma unroll
  for (int off = 1; off < 32; off <<= 1) { s += __shfl_xor(s, off); s2 += __shfl_xor(s2, off); }
  float mu  = s * (1.f / HC_);
  float var = s2 * (1.f / HC_) - mu * mu;
  float inv = rsqrtf(var + 1e-5f);
#pragma unroll
  for (int j = 0; j < 4; j++) {
    int c = lane + j * 32;
    out[(size_t)row * HC_ + c] = (v[j] - mu) * inv * g[c] + b[c];
  }
}

// ---------------- small utility / scatter kernels ----------------
__global__ void fill_f32_kernel(float* p, float v, int n) {
  int t = blockIdx.x * blockDim.x + threadIdx.x;
  if (t < n) p[t] = v;
}
__global__ void fill_u32_kernel(unsigned* p, unsigned v, int n) {
  int t = blockIdx.x * blockDim.x + threadIdx.x;
  if (t < n) p[t] = v;
}
__global__ void deg_kernel(const int* __restrict__ srcv, float* deg) {
  int t = blockIdx.x * blockDim.x + threadIdx.x;
  if (t < EDGES) unsafeAtomicAdd(deg + srcv[t], 1.0f);
}
__global__ void dis_kernel(const float* __restrict__ deg, float* dis) {
  int t = blockIdx.x * blockDim.x + threadIdx.x;
  if (t < NODES) {
    float d = deg[t];
    dis[t] = (d > 0.f) ? rsqrtf(fmaxf(d, 1.f)) : 0.f;
  }
}
__global__ void edgew_kernel(const int* __restrict__ srcv, const int* __restrict__ dstv,
                             const float* __restrict__ dis, float* wgt) {
  int t = blockIdx.x * blockDim.x + threadIdx.x;
  if (t < EDGES) wgt[t] = -(dis[srcv[t]] * dis[dstv[t]]);
}
__global__ void lap_kernel(const float* __restrict__ vin, const float* __restrict__ wgt,
                           const int* __restrict__ srcv, const int* __restrict__ dstv,
                           float* vout) {
  int t = blockIdx.x * blockDim.x + threadIdx.x;
  if (t >= EDGES * CIN_) return;
  int e = t >> 6, c = t & 63;
  unsafeAtomicAdd(vout + (size_t)dstv[e] * CIN_ + c, wgt[e] * vin[(size_t)srcv[e] * CIN_ + c]);
}
__global__ void chebnext_kernel(float* t2, const float* __restrict__ t0, int n) {
  int t = blockIdx.x * blockDim.x + threadIdx.x;
  if (t < n) t2[t] = 2.f * t2[t] - t0[t];
}

// ---- edge attention (segment softmax over dst) ----
__device__ __forceinline__ unsigned f32key(float f) {
  unsigned u = __float_as_uint(f);
  return (u & 0x80000000u) ? ~u : (u | 0x80000000u);
}
__device__ __forceinline__ float f32unkey(unsigned k) {
  unsigned u = (k & 0x80000000u) ? (k & 0x7fffffffu) : ~k;
  return __uint_as_float(u);
}
__global__ void alpha_kernel(const float* __restrict__ qb, const float* __restrict__ kb,
                             const float* __restrict__ eh,
                             const int* __restrict__ srcv, const int* __restrict__ dstv,
                             float* alpha, unsigned* smax) {
  int t = blockIdx.x * blockDim.x + threadIdx.x;
  if (t >= EDGES * HEADS_) return;
  int e = t >> 2, hd = t & 3;
  int s = srcv[e], d = dstv[e];
  const float* qp = qb + (size_t)d * HC_ + hd * HD_;
  const float* kp = kb + (size_t)s * HC_ + hd * HD_;
  const float* ep = eh + (size_t)e * HC_ + hd * HD_;
  float a = 0.f;
#pragma unroll 8
  for (int j = 0; j < HD_; j++) a += qp[j] * (kp[j] + ep[j]);
  a *= 0.17677669529663687f;
  alpha[t] = a;
  atomicMax(smax + d * HEADS_ + hd, f32key(a));
}
__global__ void alphaexp_kernel(const int* __restrict__ dstv, float* alpha,
                                const unsigned* __restrict__ smax, float* ssum) {
  int t = blockIdx.x * blockDim.x + threadIdx.x;
  if (t >= EDGES * HEADS_) return;
  int e = t >> 2, hd = t & 3;
  int d = dstv[e];
  float ex = __expf(alpha[t] - f32unkey(smax[d * HEADS_ + hd]));
  alpha[t] = ex;
  unsafeAtomicAdd(ssum + d * HEADS_ + hd, ex);
}
__global__ void msg_kernel(const float* __restrict__ vb, const float* __restrict__ eh,
                           const int* __restrict__ srcv, const int* __restrict__ dstv,
                           const float* __restrict__ alpha, const float* __restrict__ ssum,
                           float* agg) {
  int t = blockIdx.x * blockDim.x + threadIdx.x;
  if (t >= EDGES * HC_) return;
  int e = t >> 7, c = t & 127, hd = c >> 5;
  int s = srcv[e], d = dstv[e];
  float al = alpha[e * HEADS_ + hd] / (ssum[d * HEADS_ + hd] + 1e-16f);
  unsafeAtomicAdd(agg + (size_t)d * HC_ + c,
                  (vb[(size_t)s * HC_ + c] + eh[(size_t)e * HC_ + c]) * al);
}

// h += emb + sinusoidal position
__global__ void embpos_kernel(float* h, const float* __restrict__ emb) {
  int t = blockIdx.x * blockDim.x + threadIdx.x;
  if (t >= NODES * HC_) return;
  int row = t >> 7, c = t & 127, j = c & 63;
  float freq = expf((float)j * (-9.210340371976184f / 64.f));
  float ang  = (float)row * freq;
  float pos  = (c < 64) ? sinf(ang) : cosf(ang);
  h[t] += emb[t] + pos;
}

// final projection HC -> 2
__global__ void fc_kernel(const float* __restrict__ h, const float* __restrict__ W,
                          const float* __restrict__ b, float* out) {
  int t = blockIdx.x * blockDim.x + threadIdx.x;
  if (t >= NODES) return;
  float a0 = b[0], a1 = b[1];
#pragma unroll 8
  for (int j = 0; j < HC_; j++) {
    float x = h[(size_t)t * HC_ + j];
    a0 += x * W[j * 2 + 0];
    a1 += x * W[j * 2 + 1];
  }
  out[t * 2 + 0] = a0;
  out[t * 2 + 1] = a1;
}

// ---------------------------------------------------------------------------
extern "C" void kernel_launch(void* const* d_in, const int* in_sizes, int n_in,
                              void* d_out, int out_size, void* d_ws, size_t ws_size,
                              hipStream_t stream) {
  (void)in_sizes; (void)n_in; (void)out_size; (void)ws_size;
  const float* x  = (const float*)d_in[0];
  const int*   ei = (const int*)d_in[1];
  const float* ea = (const float*)d_in[2];
  const int* srcv = ei;
  const int* dstv = ei + EDGES;
  auto F = [&](int i) { return (const float*)d_in[i]; };

  // ---- workspace carve-up (floats) ----
  float* W = (float*)d_ws;
  size_t off = 0;
  auto alloc = [&](size_t n) { float* p = W + off; off += n; return p; };
  float* h     = alloc((size_t)NODES * HC_);
  float* emb   = alloc((size_t)NODES * HC_);
  float* zb    = alloc((size_t)NODES * HC_);
  float* attnb = alloc((size_t)NODES * HC_);
  float* aggb  = alloc((size_t)NODES * HC_);
  float* qb    = alloc((size_t)NODES * HC_);
  float* kb    = alloc((size_t)NODES * HC_);
  float* vb    = alloc((size_t)NODES * HC_);
  float* tmp   = alloc((size_t)NODES * 3 * HC_);
  float* deg   = alloc(NODES);
  float* dis   = alloc(NODES);
  float* tx1   = alloc((size_t)NODES * CIN_);
  float* tx2   = alloc((size_t)NODES * CIN_);
  float* tx3   = alloc((size_t)NODES * CIN_);
  float* wgt   = alloc(EDGES);
  float* alpha = alloc((size_t)EDGES * HEADS_);
  unsigned* smax = (unsigned*)alloc((size_t)NODES * HEADS_);
  float* ssum  = alloc((size_t)NODES * HEADS_);
  float* ebuf  = alloc((size_t)EDGES * HC_);
  float* ehb   = alloc((size_t)EDGES * HC_);

  auto gemm = [&](int epi, const float* A, int lda, const float* B, int ldb,
                  const float* bias, const float* R, float* C, int ldc,
                  int M, int Nc, int K) {
    int blocks = CDIV((M >> 4) * (Nc >> 4), 8);
    switch (epi) {
      case 0: gemm16<0><<<blocks, 256, 0, stream>>>(A, lda, B, ldb, bias, R, C, ldc, M, Nc, K); break;
      case 1: gemm16<1><<<blocks, 256, 0, stream>>>(A, lda, B, ldb, bias, R, C, ldc, M, Nc, K); break;
      case 3: gemm16<3><<<blocks, 256, 0, stream>>>(A, lda, B, ldb, bias, R, C, ldc, M, Nc, K); break;
      case 4: gemm16<4><<<blocks, 256, 0, stream>>>(A, lda, B, ldb, bias, R, C, ldc, M, Nc, K); break;
      default: gemm16<5><<<blocks, 256, 0, stream>>>(A, lda, B, ldb, bias, R, C, ldc, M, Nc, K); break;
    }
  };
  auto fillf = [&](float* p, float v, int n) {
    fill_f32_kernel<<<CDIV(n, 256), 256, 0, stream>>>(p, v, n);
  };
  auto lnorm = [&](const float* X, const float* Rsd, const float* g, const float* bvec, float* o) {
    ln_kernel<<<CDIV(NODES * 32, 256), 256, 0, stream>>>(X, Rsd, g, bvec, o, NODES);
  };

  // ---- emb = x @ W_enc1 ----
  gemm(0, x, CIN_, F(4), HC_, nullptr, nullptr, emb, HC_, NODES, HC_, CIN_);

  // ---- ChebConv(K=5) -> h ----
  const float* chebW = F(5);
  fillf(deg, 0.f, NODES);
  deg_kernel<<<CDIV(EDGES, 256), 256, 0, stream>>>(srcv, deg);
  dis_kernel<<<CDIV(NODES, 256), 256, 0, stream>>>(deg, dis);
  edgew_kernel<<<CDIV(EDGES, 256), 256, 0, stream>>>(srcv, dstv, dis, wgt);
  fillf(tx1, 0.f, NODES * CIN_);
  lap_kernel<<<CDIV(EDGES * CIN_, 256), 256, 0, stream>>>(x, wgt, srcv, dstv, tx1);
  gemm(0, x,   CIN_, chebW + 0 * CIN_ * HC_, HC_, nullptr, nullptr, h, HC_, NODES, HC_, CIN_);
  gemm(4, tx1, CIN_, chebW + 1 * CIN_ * HC_, HC_, nullptr, h, h, HC_, NODES, HC_, CIN_);
  fillf(tx2, 0.f, NODES * CIN_);                                      // Tx2 = 2 lap(Tx1) - Tx0
  lap_kernel<<<CDIV(EDGES * CIN_, 256), 256, 0, stream>>>(tx1, wgt, srcv, dstv, tx2);
  chebnext_kernel<<<CDIV(NODES * CIN_, 256), 256, 0, stream>>>(tx2, x, NODES * CIN_);
  gemm(4, tx2, CIN_, chebW + 2 * CIN_ * HC_, HC_, nullptr, h, h, HC_, NODES, HC_, CIN_);
  fillf(tx3, 0.f, NODES * CIN_);                                      // Tx3 = 2 lap(Tx2) - Tx1
  lap_kernel<<<CDIV(EDGES * CIN_, 256), 256, 0, stream>>>(tx2, wgt, srcv, dstv, tx3);
  chebnext_kernel<<<CDIV(NODES * CIN_, 256), 256, 0, stream>>>(tx3, tx1, NODES * CIN_);
  gemm(4, tx3, CIN_, chebW + 3 * CIN_ * HC_, HC_, nullptr, h, h, HC_, NODES, HC_, CIN_);
  fillf(tx1, 0.f, NODES * CIN_);                                      // Tx4 = 2 lap(Tx3) - Tx2
  lap_kernel<<<CDIV(EDGES * CIN_, 256), 256, 0, stream>>>(tx3, wgt, srcv, dstv, tx1);
  chebnext_kernel<<<CDIV(NODES * CIN_, 256), 256, 0, stream>>>(tx1, tx2, NODES * CIN_);
  gemm(5, tx1, CIN_, chebW + 4 * CIN_ * HC_, HC_, F(6), h, h, HC_, NODES, HC_, CIN_);

  // ---- e = edge_attr @ edge_W + edge_b ----
  gemm(1, ea, EDIM_, F(7), HC_, F(8), nullptr, ebuf, HC_, EDGES, HC_, EDIM_);

  // ---- TransformerConv (edge attention) ----
  auto tconv = [&](int base, const float* hin, float* xout) {
    gemm(1, hin, HC_, F(base + 2), HC_, F(base + 6), nullptr, qb, HC_, NODES, HC_, HC_); // Wq,bq
    gemm(1, hin, HC_, F(base + 1), HC_, F(base + 5), nullptr, kb, HC_, NODES, HC_, HC_); // Wk,bk
    gemm(1, hin, HC_, F(base + 4), HC_, F(base + 8), nullptr, vb, HC_, NODES, HC_, HC_); // Wv,bv
    gemm(0, ebuf, HC_, F(base + 0), HC_, nullptr, nullptr, ehb, HC_, EDGES, HC_, HC_);   // We
    fill_u32_kernel<<<CDIV(NODES * HEADS_, 256), 256, 0, stream>>>(smax, 0x007FFFFFu, NODES * HEADS_);
    fillf(ssum, 0.f, NODES * HEADS_);
    fillf(aggb, 0.f, NODES * HC_);
    alpha_kernel<<<CDIV(EDGES * HEADS_, 256), 256, 0, stream>>>(qb, kb, ehb, srcv, dstv, alpha, smax);
    alphaexp_kernel<<<CDIV(EDGES * HEADS_, 256), 256, 0, stream>>>(dstv, alpha, smax, ssum);
    msg_kernel<<<CDIV(EDGES * HC_, 256), 256, 0, stream>>>(vb, ehb, srcv, dstv, alpha, ssum, aggb);
    gemm(5, hin, HC_, F(base + 3), HC_, F(base + 7), aggb, xout, HC_, NODES, HC_, HC_);  // Wskip,bskip
  };

  tconv(47, h, tmp);
  lnorm(h, tmp, F(40), F(39), h);                                               // norm1
  gemm(3, h, HC_, F(35), HC_, F(36), nullptr, tmp, HC_, NODES, HC_, HC_);       // relu(lin1)
  lnorm(h, tmp, F(42), F(41), h);                                               // norm2
  tconv(56, h, tmp);
  lnorm(h, tmp, F(44), F(43), h);                                               // norm3
  gemm(3, h, HC_, F(37), HC_, F(38), nullptr, tmp, HC_, NODES, HC_, HC_);       // relu(lin2)
  lnorm(h, tmp, F(46), F(45), h);                                               // norm4
  embpos_kernel<<<CDIV(NODES * HC_, 256), 256, 0, stream>>>(h, emb);

  // ---- dense Transformer encoder layers ----
  auto enc = [&](int base) {
    lnorm(h, nullptr, F(base + 7), F(base + 6), zb);                                      // ln1
    gemm(1, zb, HC_, F(base + 4), 3 * HC_, F(base + 5), nullptr, tmp, 3 * HC_, NODES, 3 * HC_, HC_);
    attn_kernel<<<(NODES / 16) * HEADS_ / 8, 256, 0, stream>>>(tmp, attnb);
    gemm(5, attnb, HC_, F(base + 10), HC_, F(base + 11), h, h, HC_, NODES, HC_, HC_);     // out proj
    lnorm(h, nullptr, F(base + 9), F(base + 8), zb);                                      // ln2
    gemm(3, zb, HC_, F(base + 0), FF_, F(base + 1), nullptr, tmp, FF_, NODES, FF_, HC_);  // relu(ff1)
    gemm(5, tmp, FF_, F(base + 2), HC_, F(base + 3), h, h, HC_, NODES, HC_, FF_);         // ff2
  };
  enc(9);
  enc(21);

  fc_kernel<<<CDIV(NODES, 256), 256, 0, stream>>>(h, F(33), F(34), (float*)d_out);
}